// NoiseGenerator_13400297964158
// MI455X (gfx1250) — compile-verified
//
#include <hip/hip_runtime.h>
#include <hip/hip_bf16.h>

typedef float v2f __attribute__((ext_vector_type(2)));
typedef float v8f __attribute__((ext_vector_type(8)));

constexpr int SEG_LEN    = 1024;              // output samples per segment (column)
constexpr int WARM       = 3072;              // warm-up samples (0.99^3072 ~ 4e-14)
constexpr int TILES      = (WARM + SEG_LEN) / 16;   // 256 tiles per segment
constexpr int WARM_TILES = WARM / 16;               // 192 warm-up tiles

__device__ __forceinline__ float powi_f(float c, int e) {
    float p = 1.0f;
    for (int i = 0; i < e; ++i) p *= c;
    return p;
}

// Cascade:  y_lp[n] = clp*y_lp[n-1] + (1-clp)*x[n]
//           y_hp[n] = chp*y_hp[n-1] + chp*(y_lp[n] - y_lp[n-1])
// Over a 16-sample tile with carried states (carryLp = y_lp[-1], carryHp = y_hp[-1]):
//   y_hp[m] = sum_k T3[m][k]*x[k] + qlp[m]*carryLp + chp^(m+1)*carryHp
// where dT[r][k] maps x -> d[r]=y_lp[r]-y_lp[r-1]:
//   dT[r][r] = (1-clp);  dT[r][k<r] = -(1-clp)^2 * clp^(r-1-k)
//   T3[m][k] = sum_{r=k..m} chp^(m-r+1) * dT[r][k]
//   qlp[m]   = (clp-1) * sum_{r=0..m} chp^(m-r+1) * clp^r
__global__ __launch_bounds__(256) void
noisegen_iir_wmma(const float* __restrict__ prm,
                  const float* __restrict__ x,
                  float* __restrict__ out, int n)
{
    const int lane = threadIdx.x & 31;
    const int col  = lane & 15;   // N column (segment within wave)
    const int h    = lane >> 4;   // lane half
    const int wave = (blockIdx.x * blockDim.x + threadIdx.x) >> 5;

    if (wave * 16 * SEG_LEN >= n) return;   // wave-uniform exit: EXEC stays all-ones

    const int segStart = (wave * 16 + col) * SEG_LEN;
    const int base     = segStart - WARM;   // x[idx<0] treated as 0 == exact zero state

    // ---- clamped parameters ----
    const float attack = fmaxf(prm[0], 1e-7f);
    const float decay  = fmaxf(prm[1], 1e-7f);
    const float clp    = fminf(fmaxf(prm[2], 1e-7f), 0.99f);
    const float chp    = fminf(fmaxf(prm[3], 1e-7f), 0.99f);
    const float gain   = fmaxf(prm[4], 1e-7f);
    const float inv_nm1 = 1.0f / (float)(n - 1);
    const float ia = -1.0f / attack;
    const float id = -1.0f / decay;

    auto T3f = [&](int m, int k) -> float {   // init-only helper
        if (k > m) return 0.0f;
        float s = 0.0f;
        for (int r = k; r <= m; ++r) {
            float dT = (r == k) ? (1.0f - clp)
                                : -(1.0f - clp) * (1.0f - clp) * powi_f(clp, r - 1 - k);
            s += powi_f(chp, m - r + 1) * dT;
        }
        return s;
    };

    // ---- per-lane constant matrices / weights ----
    // A-layout (16x4 f32, 2 VGPRs per K-chunk): reg q = 2*kc + s  holds  k = 4*kc + s + 2*h, m = col
    float A3[8], wL[8], wH[8];
    for (int q = 0; q < 8; ++q) {
        const int k = 4 * (q >> 1) + (q & 1) + 2 * h;
        A3[q] = T3f(col, k);                              // HP-from-X matrix (A operand)
        wL[q] = (1.0f - clp) * powi_f(clp, 15 - k);       // row y_lp[15] weights (carryLp dot)
        wH[q] = T3f(15, k);                               // row y_hp[15] weights (carryHp dot)
    }
    // C-layout: reg r holds m = r + 8*h
    float Qc[8], Pc[8];
    for (int r = 0; r < 8; ++r) {
        const int m = r + 8 * h;
        Pc[r] = powi_f(chp, m + 1);
        float s = 0.0f;
        for (int r2 = 0; r2 <= m; ++r2) s += powi_f(chp, m - r2 + 1) * powi_f(clp, r2);
        Qc[r] = (clp - 1.0f) * s;
    }
    const float aLL = powi_f(clp, 16);   // carryLp decay per tile
    const float bHH = powi_f(chp, 16);   // carryHp decay per tile
    float q15;
    {
        float s = 0.0f;
        for (int r2 = 0; r2 <= 15; ++r2) s += powi_f(chp, 16 - r2) * powi_f(clp, r2);
        q15 = (clp - 1.0f) * s;          // carryLp -> carryHp coupling per tile
    }

    // ---- B tile loader: 4 x float2 per lane; B chunk kc reg s holds x[base+16i+4kc+s+2h] of column `col`
    auto loadTile = [&](int i, v2f* Bv) {
        const int t0 = base + i * 16 + 2 * h;
#pragma unroll
        for (int kc = 0; kc < 4; ++kc) {
            const int p = t0 + 4 * kc;          // always even -> 8B aligned
            v2f v = {0.0f, 0.0f};
            if (p >= 0 && p + 1 < n) {
                const float2 w = *(const float2*)(x + p);
                v.x = w.x; v.y = w.y;
            } else {
                if (p >= 0 && p < n)         v.x = x[p];
                if (p + 1 >= 0 && p + 1 < n) v.y = x[p + 1];
            }
            Bv[kc] = v;
        }
    };

    float carryLp = 0.0f, carryHp = 0.0f;
    v2f B[4];
    loadTile(0, B);

    for (int i = 0; i < TILES; ++i) {
        v2f Bn[4];
        if (i + 1 < TILES) loadTile(i + 1, Bn);   // prefetch next tile (off critical path)

        // --- carry row-dots: each half holds 8 of the 16 k's of its column ---
        float dl = 0.0f, dh = 0.0f;
#pragma unroll
        for (int kc = 0; kc < 4; ++kc) {
            dl = fmaf(wL[2 * kc], B[kc].x, dl); dl = fmaf(wL[2 * kc + 1], B[kc].y, dl);
            dh = fmaf(wH[2 * kc], B[kc].x, dh); dh = fmaf(wH[2 * kc + 1], B[kc].y, dh);
        }
        dl += __shfl_xor(dl, 16);   // combine lane halves -> full 16-tap dot in every lane
        dh += __shfl_xor(dh, 16);

        if (i >= WARM_TILES) {      // uniform predicate: EXEC all-ones around WMMA
            // seed C with carried-state contributions, then Y_hp = T3*X + C
            v8f C;
#pragma unroll
            for (int r = 0; r < 8; ++r)
                C[r] = fmaf(Qc[r], carryLp, Pc[r] * carryHp);
#pragma unroll
            for (int kc = 0; kc < 4; ++kc) {
                v2f a = { A3[2 * kc], A3[2 * kc + 1] };
                C = __builtin_amdgcn_wmma_f32_16x16x4_f32(
                        false, a, false, B[kc], (short)0, C, false, false);
            }
            // C reg r = sample m = r + 8*h of column `col` -> 8 contiguous outputs per lane
            const int idx0 = base + i * 16 + 8 * h;
            float o[8];
#pragma unroll
            for (int r = 0; r < 8; ++r) {
                const float t = (float)(idx0 + r) * inv_nm1;
                const float env = (1.0f - __expf(t * ia)) * __expf(t * id) * gain;
                o[r] = C[r] * env;
            }
            if (idx0 >= 0 && idx0 + 7 < n) {
                *(float4*)(out + idx0)     = make_float4(o[0], o[1], o[2], o[3]);
                *(float4*)(out + idx0 + 4) = make_float4(o[4], o[5], o[6], o[7]);
            } else {
#pragma unroll
                for (int r = 0; r < 8; ++r) {
                    const int g = idx0 + r;
                    if (g >= 0 && g < n) out[g] = o[r];
                }
            }
        }

        // --- state update (uses OLD carryLp for the coupling term; order matters) ---
        carryHp = fmaf(bHH, carryHp, fmaf(q15, carryLp, dh));
        carryLp = fmaf(aLL, carryLp, dl);

#pragma unroll
        for (int kc = 0; kc < 4; ++kc) B[kc] = Bn[kc];
    }
}

extern "C" void kernel_launch(void* const* d_in, const int* in_sizes, int n_in,
                              void* d_out, int out_size, void* d_ws, size_t ws_size,
                              hipStream_t stream) {
    const float* prm   = (const float*)d_in[0];   // 5 params
    const float* noise = (const float*)d_in[1];   // n samples
    float* out = (float*)d_out;
    const int n = in_sizes[1];

    const int segments       = (n + SEG_LEN - 1) / SEG_LEN;
    const int waves          = (segments + 15) / 16;
    const int threads        = 256;                 // 8 wave32 per block
    const int wavesPerBlock  = threads / 32;
    const int blocks         = (waves + wavesPerBlock - 1) / wavesPerBlock;

    noisegen_iir_wmma<<<blocks, threads, 0, stream>>>(prm, noise, out, n);
}